// ConcatHandshaking_76759655514537
// MI455X (gfx1250) — compile-verified
//
#include <hip/hip_runtime.h>
#include <hip/hip_bf16.h>
#include <math.h>

typedef __attribute__((ext_vector_type(2))) float v2f;
typedef __attribute__((ext_vector_type(4))) float v4f;
typedef __attribute__((ext_vector_type(8))) float v8f;

#define B_ 16
#define S_ 128
#define D_ 768
#define H_ 768
#define M_ (B_ * S_)              /* 2048 rows of the flattened GEMM */
#define P_ ((S_ * (S_ + 1)) / 2)  /* 8256 upper-triangular pairs */

#if defined(__has_builtin)
#if __has_builtin(__builtin_amdgcn_tanhf)
#define TANH_F32(x) __builtin_amdgcn_tanhf(x)
#endif
#endif
#ifndef TANH_F32
#define TANH_F32(x) tanhf(x)
#endif

// ---------------------------------------------------------------------------
// Kernel 1: proj[z][m][n] = sum_k hidden[m][k] * W[z*D + k][n]  (+ b[n] if z==1)
// One wave computes a 16x64 output tile using V_WMMA_F32_16X16X4_F32.
// Tiles: z(2) x mt(128) x nt(12) = 3072; 8 waves per 256-thread block -> 384 blocks.
// ---------------------------------------------------------------------------
__global__ __launch_bounds__(256) void gemm_proj_kernel(
    const float* __restrict__ A,     // hidden, (M_, D_) row-major
    const float* __restrict__ W,     // (2*D_, H_) row-major
    const float* __restrict__ bias,  // (H_)
    float* __restrict__ proj)        // (2, M_, H_)
{
    const int lane = threadIdx.x & 31;
    const int wave = threadIdx.x >> 5;
    const int tile = blockIdx.x * 8 + wave;      // 0..3071

    const int z   = tile / (128 * 12);
    const int rem = tile - z * (128 * 12);
    const int nt  = rem / 128;
    const int mt  = rem - nt * 128;

    const int m0 = mt * 16;
    const int n0 = nt * 64;
    const float* __restrict__ Wz = W + (size_t)z * D_ * H_;

    // A-fragment mapping (ISA 7.12.2, 32-bit A 16x4): lanes 0-15 -> K=0,1;
    // lanes 16-31 -> K=2,3; lane%16 = M row.  B mirrors with lane%16 = N col.
    const int row  = lane & 15;
    const int ksel = (lane >> 4) << 1;  // 0 or 2
    const int col  = lane & 15;

    v8f acc0 = {}; v8f acc1 = {}; v8f acc2 = {}; v8f acc3 = {};

    const float* __restrict__ Arow = A + (size_t)(m0 + row) * D_ + ksel;
    for (int k = 0; k < D_; k += 4) {
        v2f a;
        a.x = Arow[k];
        a.y = Arow[k + 1];

        const float* __restrict__ Wk = Wz + (size_t)(k + ksel) * H_ + n0 + col;
        v2f b0, b1, b2, b3;
        b0.x = Wk[0];       b0.y = Wk[H_];
        b1.x = Wk[16];      b1.y = Wk[H_ + 16];
        b2.x = Wk[32];      b2.y = Wk[H_ + 32];
        b3.x = Wk[48];      b3.y = Wk[H_ + 48];

        acc0 = __builtin_amdgcn_wmma_f32_16x16x4_f32(false, a, false, b0, (short)0, acc0, false, false);
        acc1 = __builtin_amdgcn_wmma_f32_16x16x4_f32(false, a, false, b1, (short)0, acc1, false, false);
        acc2 = __builtin_amdgcn_wmma_f32_16x16x4_f32(false, a, false, b2, (short)0, acc2, false, false);
        acc3 = __builtin_amdgcn_wmma_f32_16x16x4_f32(false, a, false, b3, (short)0, acc3, false, false);
    }

    float bv0 = 0.f, bv1 = 0.f, bv2 = 0.f, bv3 = 0.f;
    if (z == 1) {
        bv0 = bias[n0 + col];
        bv1 = bias[n0 + 16 + col];
        bv2 = bias[n0 + 32 + col];
        bv3 = bias[n0 + 48 + col];
    }

    // C/D layout: lane%16 = N; VGPR r holds M = r (lanes 0-15) or r+8 (lanes 16-31)
    const int mbase = m0 + ((lane >> 4) << 3);
    float* __restrict__ dst = proj + (size_t)z * M_ * H_;
    #pragma unroll
    for (int r = 0; r < 8; ++r) {
        size_t o = (size_t)(mbase + r) * H_ + n0 + col;
        dst[o]      = acc0[r] + bv0;
        dst[o + 16] = acc1[r] + bv1;
        dst[o + 32] = acc2[r] + bv2;
        dst[o + 48] = acc3[r] + bv3;
    }
}

// ---------------------------------------------------------------------------
// Kernel 2: out[b, p, :] = tanh(proj_i[b, i(p), :] + proj_j[b, j(p), :])
// One 192-thread block per (b, p) row; one v4f per thread; NT store (406 MB stream).
// ---------------------------------------------------------------------------
__global__ __launch_bounds__(192) void pair_tanh_kernel(
    const float* __restrict__ proj,  // (2, M_, H_)
    float* __restrict__ out)         // (B_, P_, H_)
{
    const int bp = blockIdx.x;       // 0 .. B_*P_-1
    const int b  = bp / P_;
    const int p  = bp - b * P_;

    // Row-major triu order: C(i) = i*S - i*(i-1)/2 pairs precede row i.
    const int t = 2 * S_ + 1;                       // 257
    int i = (int)((t - sqrtf((float)(t * t - 8 * p))) * 0.5f);
#define CROW(x) ((x) * S_ - ((x) * ((x) - 1)) / 2)
    while (i > 0 && CROW(i) > p) --i;
    while (CROW(i + 1) <= p) ++i;
    const int j = i + (p - CROW(i));
#undef CROW

    const v4f* __restrict__ pi =
        (const v4f*)(proj + ((size_t)b * S_ + i) * H_);
    const v4f* __restrict__ pj =
        (const v4f*)(proj + (size_t)M_ * H_ + ((size_t)b * S_ + j) * H_);
    v4f* __restrict__ dst = (v4f*)(out + (size_t)bp * H_);

    const int t4 = threadIdx.x;      // 0..191 == H_/4
    v4f x = pi[t4];
    v4f y = pj[t4];
    v4f r;
    r.x = TANH_F32(x.x + y.x);
    r.y = TANH_F32(x.y + y.y);
    r.z = TANH_F32(x.z + y.z);
    r.w = TANH_F32(x.w + y.w);
    __builtin_nontemporal_store(r, &dst[t4]);
}

extern "C" void kernel_launch(void* const* d_in, const int* in_sizes, int n_in,
                              void* d_out, int out_size, void* d_ws, size_t ws_size,
                              hipStream_t stream) {
    (void)in_sizes; (void)n_in; (void)out_size; (void)ws_size;
    const float* hidden = (const float*)d_in[0];  // (B,S,D) f32
    const float* W      = (const float*)d_in[1];  // (2D,H) f32
    const float* bias   = (const float*)d_in[2];  // (H,)  f32
    float* out  = (float*)d_out;                  // (B,P,H) f32
    float* proj = (float*)d_ws;                   // 2 * M_ * H_ floats = 12.6 MB

    gemm_proj_kernel<<<dim3(384), dim3(256), 0, stream>>>(hidden, W, bias, proj);
    pair_tanh_kernel<<<dim3(B_ * P_), dim3(192), 0, stream>>>(proj, out);
}